// StreamingRhythmProjector_77747497992328
// MI455X (gfx1250) — compile-verified
//
#include <hip/hip_runtime.h>
#include <stdint.h>
#include <math.h>

#define U_COLS 8192
#define TPB    256
#define V4PT   (U_COLS / (4 * TPB))   // 8 float4 per thread
#define VPT    (V4PT * 4)             // 32 elements per thread
#define NWAVE  (TPB / 32)             // 8 waves (wave32)

#if defined(__gfx1250__) && __has_builtin(__builtin_amdgcn_tensor_load_to_lds)
#define HAVE_TDM 1
#else
#define HAVE_TDM 0
#endif

typedef unsigned int u32x4_t __attribute__((ext_vector_type(4)));
typedef int          i32x4_t __attribute__((ext_vector_type(4)));
typedef int          i32x8_t __attribute__((ext_vector_type(8)));

__device__ __forceinline__ float wave_sum_f(float v) {
#pragma unroll
  for (int m = 16; m >= 1; m >>= 1) v += __shfl_xor(v, m, 32);
  return v;
}
__device__ __forceinline__ int wave_sum_i(int v) {
#pragma unroll
  for (int m = 16; m >= 1; m >>= 1) v += __shfl_xor(v, m, 32);
  return v;
}

__global__ __launch_bounds__(TPB)
void srp_row_kernel(const float* __restrict__ pause_w,
                    const float* __restrict__ boundary_s,
                    const int*   __restrict__ unit_mask,
                    const float* __restrict__ budget,
                    const float* __restrict__ prev_exec,
                    const int*   __restrict__ frontier,
                    float*       __restrict__ out)
{
  __shared__ float    lds_prev[U_COLS];          // 32 KB staged row of previous_pause_exec
  __shared__ unsigned s_hist[256];
  __shared__ int      s_ipart[2 * NWAVE];
  __shared__ float    s_fpart[2 * NWAVE];
  __shared__ unsigned s_selbin, s_newk;
  __shared__ int      s_visible, s_tslots;
  __shared__ float    s_prefix, s_ttotal;

  const int    row    = blockIdx.x;
  const int    t      = threadIdx.x;
  const int    wid    = t >> 5;
  const int    lane   = t & 31;
  const size_t rowOff = (size_t)row * U_COLS;

  // ---- Phase 0: async-stage previous_pause_exec row into LDS (Tensor Data Mover) ----
#if HAVE_TDM
  if (wid == 0) {
    unsigned lds_off = (unsigned)(size_t)(void*)&lds_prev[0];            // LDS byte offset (addr[31:0])
    unsigned long long ga = (unsigned long long)(const void*)(prev_exec + rowOff);
    // D# group0: count=1 | lds_addr | global_addr[56:0] | type=2 (bits 127:126)
    u32x4_t g0 = { 1u,
                   lds_off,
                   (unsigned)ga,
                   ((unsigned)(ga >> 32) & 0x01FFFFFFu) | 0x80000000u };
    // D# group1: data_size=2 (4B); tensor_dim0=8192; tensor_dim1=1; tile_dim0=8192;
    //            tile_dim1=1; tensor_dim0_stride=8192; workgroup_mask=0 (not in cluster)
    i32x8_t g1 = { (int)0x00020000u,
                   (int)(U_COLS << 16),   // tensor_dim0[15:0] @ bits 63:48
                   (int)(1u << 16),       // tensor_dim1[15:0]=1 @ bits 95:80
                   (int)(U_COLS << 16),   // tile_dim0 @ bits 127:112
                   1,                     // tile_dim1=1, tile_dim2=0
                   U_COLS,                // tensor_dim0_stride low32
                   0, 0 };
    i32x4_t gz = { 0, 0, 0, 0 };
#if __clang_major__ >= 23
    i32x8_t gz8 = { 0, 0, 0, 0, 0, 0, 0, 0 };
    __builtin_amdgcn_tensor_load_to_lds(g0, g1, gz, gz, gz8, 0);
#else
    __builtin_amdgcn_tensor_load_to_lds(g0, g1, gz, gz, 0);
#endif
  }
#else
  { // fallback: cooperative copy
    const float4* pv4 = (const float4*)(prev_exec + rowOff);
    float4* lp4 = (float4*)lds_prev;
#pragma unroll
    for (int e = 0; e < V4PT; ++e) lp4[e * TPB + t] = pv4[e * TPB + t];
  }
#endif

  // ---- Phase A: single global read of pause/boundary/mask; keys kept in registers ----
  const float4* pw4 = (const float4*)(pause_w    + rowOff);
  const float4* bs4 = (const float4*)(boundary_s + rowOff);
  const int4*   mk4 = (const int4*)(unit_mask    + rowOff);

  int fr = frontier[row];
  fr = fr < 0 ? 0 : (fr > U_COLS ? U_COLS : fr);

  unsigned key[VPT];
  int myVis = 0, myTS = 0;

#pragma unroll
  for (int e = 0; e < V4PT; ++e) {
    const int f = e * TPB + t;          // float4 index in row
    float4 p = pw4[f];
    float4 b = bs4[f];
    int4   m = mk4[f];
    const float pc[4] = { p.x, p.y, p.z, p.w };
    const float bc[4] = { b.x, b.y, b.z, b.w };
    const int   mc[4] = { m.x, m.y, m.z, m.w };
    const int j0 = f * 4;
#pragma unroll
    for (int c = 0; c < 4; ++c) {
      const int  j     = j0 + c;
      const bool valid = (mc[c] != 0);
      const float gain = 1.0f + 0.15f * (0.1f + fmaxf(bc[c], 0.0f));
      float r = fmaxf(pc[c], 0.0f) * gain;
      r = fmaxf(r, 0.0f);               // kill any -0.0 so bit order == value order
      key[e * 4 + c] = valid ? (__float_as_uint(r) + 1u) : 0u;
      myVis += valid ? 1 : 0;
      myTS  += (valid && j >= fr) ? 1 : 0;
    }
  }

  // block reductions: visible count, tail slot count (deterministic order)
  {
    int wv = wave_sum_i(myVis);
    int wt = wave_sum_i(myTS);
    if (lane == 0) { s_ipart[wid] = wv; s_ipart[NWAVE + wid] = wt; }
  }
  __syncthreads();
  if (t == 0) {
    int v = 0, ts = 0;
    for (int i = 0; i < NWAVE; ++i) { v += s_ipart[i]; ts += s_ipart[NWAVE + i]; }
    s_visible = v; s_tslots = ts;
  }
  __syncthreads();

  const int visible = s_visible;
  const int tslots  = s_tslots;
  int tk = (int)rintf((float)visible * 0.35f);   // rndne == jnp.round
  if (visible > 0) tk = max(tk, 1);
  tk = min(tk, visible);
  const bool full_keep = (tk >= visible);

  // ---- Phase S: exact tk-th largest via 4-pass MSD radix select over register keys ----
  unsigned key_sel = 0;
  if (!full_keep) {
    unsigned kth = (unsigned)tk;
    unsigned pfx = 0;
#pragma unroll 1
    for (int shift = 24; shift >= 0; shift -= 8) {
      const unsigned himask = (shift == 24) ? 0u : ~((1u << (shift + 8)) - 1u);
      s_hist[t] = 0;
      __syncthreads();
#pragma unroll
      for (int i = 0; i < VPT; ++i) {
        const unsigned kk = key[i];
        if ((kk & himask) == pfx)
          atomicAdd(&s_hist[(kk >> shift) & 0xFFu], 1u);
      }
      __syncthreads();
      // suffix-sum: s_hist[b] = #keys with byte >= b (matching prefix)
      for (int off = 1; off < 256; off <<= 1) {
        const unsigned v = s_hist[t];
        const unsigned a = (t + off < 256) ? s_hist[t + off] : 0u;
        __syncthreads();
        s_hist[t] = v + a;
        __syncthreads();
      }
      const unsigned ge = s_hist[t];
      const unsigned gt = (t < 255) ? s_hist[t + 1] : 0u;
      if (ge >= kth && gt < kth) { s_selbin = (unsigned)t; s_newk = kth - gt; }
      __syncthreads();
      pfx |= (s_selbin << shift);
      kth  = s_newk;
      __syncthreads();
    }
    key_sel = pfx;
  }
  const float thr = full_keep ? 0.0f : __uint_as_float(key_sel - 1u);

  // ---- wait for TDM-staged prev row, release to all waves ----
#if HAVE_TDM
  if (wid == 0) {
#if __has_builtin(__builtin_amdgcn_s_wait_tensorcnt)
    __builtin_amdgcn_s_wait_tensorcnt(0);
#else
    asm volatile("s_wait_tensorcnt 0" ::: "memory");
#endif
  }
#endif
  __syncthreads();

  // ---- Phase B: prefix-budget sum and tail total ----
  float myPref = 0.0f, myTT = 0.0f;
#pragma unroll
  for (int i = 0; i < VPT; ++i) {
    const int e = i >> 2, c = i & 3;
    const int j = (e * TPB + t) * 4 + c;
    const unsigned kk = key[i];
    if (kk != 0u) {
      if (j < fr) {
        myPref += lds_prev[j];
      } else {
        const float r = __uint_as_float(kk - 1u);
        const float gate = full_keep ? 1.0f
                         : (1.0f / (1.0f + expf((thr - r) / 0.12f)));
        myTT += r * gate;                 // tail_cand (>= 0)
      }
    }
  }
  {
    float wp  = wave_sum_f(myPref);
    float wtt = wave_sum_f(myTT);
    if (lane == 0) { s_fpart[wid] = wp; s_fpart[NWAVE + wid] = wtt; }
  }
  __syncthreads();
  if (t == 0) {
    float a = 0.0f, b = 0.0f;
    for (int i = 0; i < NWAVE; ++i) { a += s_fpart[i]; b += s_fpart[NWAVE + i]; }
    s_prefix = a; s_ttotal = b;
  }
  __syncthreads();

  const float prefix_sum = s_prefix;
  const float tail_total = s_ttotal;
  const float remaining  = fmaxf(budget[row] - prefix_sum, 0.0f);
  const bool  use_fb     = !(tail_total > 1e-6f);
  const float invTT      = 1.0f / fmaxf(tail_total, 1e-6f);
  const float fb         = (tslots > 0) ? (1.0f / fmaxf((float)tslots, 1.0f)) : 0.0f;

  // ---- Phase C: write output (prefix reuse + proportional tail distribution) ----
  float* outRow = out + rowOff;
#pragma unroll
  for (int e = 0; e < V4PT; ++e) {
    const int f = e * TPB + t;
    float oc[4];
#pragma unroll
    for (int c = 0; c < 4; ++c) {
      const int i = e * 4 + c;
      const int j = f * 4 + c;
      const unsigned kk = key[i];
      float v = 0.0f;
      if (kk != 0u) {
        if (j < fr) {
          v = lds_prev[j];
        } else {
          const float r = __uint_as_float(kk - 1u);
          const float gate = full_keep ? 1.0f
                           : (1.0f / (1.0f + expf((thr - r) / 0.12f)));
          const float tc   = r * gate;
          const float dist = use_fb ? fb : tc * invTT;
          v = dist * remaining;
        }
      }
      oc[c] = v;
    }
    float4 o; o.x = oc[0]; o.y = oc[1]; o.z = oc[2]; o.w = oc[3];
    ((float4*)outRow)[f] = o;
  }
}

extern "C" void kernel_launch(void* const* d_in, const int* in_sizes, int n_in,
                              void* d_out, int out_size, void* d_ws, size_t ws_size,
                              hipStream_t stream) {
  const float* pause_w    = (const float*)d_in[0];
  const float* boundary_s = (const float*)d_in[1];
  const int*   unit_mask  = (const int*)d_in[2];
  const float* budget     = (const float*)d_in[3];
  const float* prev_exec  = (const float*)d_in[4];
  const int*   frontier   = (const int*)d_in[5];
  float*       out        = (float*)d_out;

  const int B = in_sizes[3];   // pause_budget_win is [B,1] -> B rows
  srp_row_kernel<<<B, TPB, 0, stream>>>(pause_w, boundary_s, unit_mask,
                                        budget, prev_exec, frontier, out);
  (void)n_in; (void)out_size; (void)d_ws; (void)ws_size;
}